// DANet_Attention_1520418422915
// MI455X (gfx1250) — compile-verified
//
#include <hip/hip_runtime.h>
#include <hip/hip_bf16.h>

typedef __attribute__((ext_vector_type(2))) float v2f;
typedef __attribute__((ext_vector_type(8))) float v8f;

// D = A(16x4) * B(4x16) + C, fp32 WMMA (CDNA5 V_WMMA_F32_16X16X4_F32)
__device__ inline v8f wmma_f32(v2f a, v2f b, v8f c) {
  return __builtin_amdgcn_wmma_f32_16x16x4_f32(false, a, false, b, (short)0, c,
                                               false, false);
}

// ---------------------------------------------------------------------------
// QKV projection: out[o][n] = sum_c W[o][c] * x[b][c][n] + bias[o]
// Stored with the reshape remap: out[(o*8 + n/1024)*1024 + (n%1024)]
// M = 16 (q,k) or 128 (v). Block = 256 threads (8 waves).
// Waves tile: msub = w % (M/16), nsub = w / (M/16); block covers M x (2048/M).
// ---------------------------------------------------------------------------
__global__ __launch_bounds__(256) void proj_kernel(
    const float* __restrict__ x,     // [B][128][8192]
    const float* __restrict__ W,     // [M][128]
    const float* __restrict__ bias,  // [M]
    float* __restrict__ out,         // per batch: M*8192 floats (remapped)
    int M) {
  extern __shared__ float Ws[];  // [16][M+1] transposed W chunk
  const int b = blockIdx.y;
  const int ncols = 2048 / M;
  const int n0 = blockIdx.x * ncols;
  const int tid = threadIdx.x;
  const int lane = tid & 31;
  const int w = tid >> 5;
  const int l16 = lane & 15;
  const int khalf = (lane >> 4) * 2;  // 0 or 2
  const int msubs = M >> 4;
  const int msub = w % msubs;
  const int nsub = w / msubs;
  const int ldw = M + 1;
  const float* xb = x + (size_t)b * 128 * 8192;
  float* outb = out + (size_t)b * M * 8192;

  v8f acc = {0.f, 0.f, 0.f, 0.f, 0.f, 0.f, 0.f, 0.f};
  const int ncol = n0 + nsub * 16;

  for (int k0 = 0; k0 < 128; k0 += 16) {
    __syncthreads();
    for (int idx = tid; idx < M * 16; idx += 256) {
      int kk = idx & 15;
      int m = idx >> 4;
      Ws[kk * ldw + m] = W[m * 128 + k0 + kk];
    }
    __syncthreads();
#pragma unroll
    for (int kl = 0; kl < 4; ++kl) {
      int ks = kl * 4 + khalf;
      v2f a, bb;
      a.x = Ws[ks * ldw + msub * 16 + l16];
      a.y = Ws[(ks + 1) * ldw + msub * 16 + l16];
      bb.x = xb[(size_t)(k0 + ks) * 8192 + ncol + l16];
      bb.y = xb[(size_t)(k0 + ks + 1) * 8192 + ncol + l16];
      acc = wmma_f32(a, bb, acc);
    }
  }
  const int rbase = msub * 16 + ((lane >> 4) << 3);
#pragma unroll
  for (int i = 0; i < 8; ++i) {
    int o = rbase + i;
    int n = ncol + l16;
    int t = n >> 10;
    int hw = n & 1023;
    outb[(size_t)((o << 3) + t) * 1024 + hw] = acc[i] + bias[o];
  }
}

// ---------------------------------------------------------------------------
// Generic batched f32 WMMA GEMM: D[m][n] (+)= alpha * sum_k A(m,k)*B(k,n) + aux
// A_MK: A stored [M][K] (else [K][M]); B_NK: B stored [N][K] (else [K][N]).
// 64x128 tile per block, TK=16 chunks staged in LDS (transpose + pad).
// 8 waves: wave w -> msub = w&3, four n-subtiles at (w>>2)*4 (+0..3).
// Each A fragment feeds 4 WMMAs; next K-chunk prefetched to WGP$.
// ---------------------------------------------------------------------------
template <bool A_MK, bool B_NK, bool ACCUM>
__global__ __launch_bounds__(256) void gemm64_kernel(
    const float* __restrict__ A, const float* __restrict__ Bm,
    float* __restrict__ D, const float* __restrict__ aux,
    const float* __restrict__ alphaPtr, int K, int lda, int ldb, int ldd,
    size_t sA, size_t sB, size_t sD, size_t sAux) {
  __shared__ float As[16 * 65];
  __shared__ float Bs[16 * 129];
  const int b = blockIdx.z;
  const float* Ab = A + (size_t)b * sA;
  const float* Bb = Bm + (size_t)b * sB;
  float* Db = D + (size_t)b * sD;
  const float* auxb = aux ? aux + (size_t)b * sAux : nullptr;
  const int m0 = blockIdx.y * 64;
  const int n0 = blockIdx.x * 128;
  const int tid = threadIdx.x;
  const int lane = tid & 31;
  const int w = tid >> 5;
  const int l16 = lane & 15;
  const int khalf = (lane >> 4) * 2;
  const int msub = w & 3;
  const int ng = w >> 2;

  v8f c[4];
#pragma unroll
  for (int j = 0; j < 4; ++j) c[j] = (v8f){0.f, 0.f, 0.f, 0.f, 0.f, 0.f, 0.f, 0.f};

  for (int k0 = 0; k0 < K; k0 += 16) {
    __syncthreads();
    if (A_MK) {
#pragma unroll
      for (int p = 0; p < 4; ++p) {
        int idx = tid + p * 256;
        int kk = idx & 15, m = idx >> 4;
        As[kk * 65 + m] = Ab[(size_t)(m0 + m) * lda + k0 + kk];
      }
    } else {
#pragma unroll
      for (int p = 0; p < 4; ++p) {
        int idx = tid + p * 256;
        int m = idx & 63, kk = idx >> 6;
        As[kk * 65 + m] = Ab[(size_t)(k0 + kk) * lda + m0 + m];
      }
    }
    if (B_NK) {
#pragma unroll
      for (int p = 0; p < 8; ++p) {
        int idx = tid + p * 256;
        int kk = idx & 15, n = idx >> 4;
        Bs[kk * 129 + n] = Bb[(size_t)(n0 + n) * ldb + k0 + kk];
      }
    } else {
#pragma unroll
      for (int p = 0; p < 8; ++p) {
        int idx = tid + p * 256;
        int n = idx & 127, kk = idx >> 7;
        Bs[kk * 129 + n] = Bb[(size_t)(k0 + kk) * ldb + n0 + n];
      }
    }
    // Pull next K-chunk toward the WGP while this chunk computes.
    if (k0 + 16 < K) {
      int kn = k0 + 16;
      if (A_MK)
        __builtin_prefetch(&Ab[(size_t)(m0 + (tid >> 4)) * lda + kn + (tid & 15)], 0, 1);
      else
        __builtin_prefetch(&Ab[(size_t)(kn + (tid >> 6)) * lda + m0 + (tid & 63)], 0, 1);
      if (B_NK)
        __builtin_prefetch(&Bb[(size_t)(n0 + (tid >> 4)) * ldb + kn + (tid & 15)], 0, 1);
      else
        __builtin_prefetch(&Bb[(size_t)(kn + (tid >> 7)) * ldb + n0 + (tid & 127)], 0, 1);
    }
    __syncthreads();
#pragma unroll
    for (int kl = 0; kl < 4; ++kl) {
      int ks = kl * 4 + khalf;
      v2f a;
      a.x = As[ks * 65 + msub * 16 + l16];
      a.y = As[(ks + 1) * 65 + msub * 16 + l16];
#pragma unroll
      for (int j = 0; j < 4; ++j) {
        int nc = (ng * 4 + j) * 16;
        v2f bb;
        bb.x = Bs[ks * 129 + nc + l16];
        bb.y = Bs[(ks + 1) * 129 + nc + l16];
        c[j] = wmma_f32(a, bb, c[j]);
      }
    }
  }
  const float alpha = alphaPtr ? alphaPtr[0] : 1.0f;
#pragma unroll
  for (int j = 0; j < 4; ++j) {
    int ncol = n0 + (ng * 4 + j) * 16 + l16;
#pragma unroll
    for (int i = 0; i < 8; ++i) {
      int row = m0 + msub * 16 + ((lane >> 4) << 3) + i;
      size_t idx = (size_t)row * ldd + ncol;
      float v = alpha * c[j][i];
      if (auxb) v += auxb[idx];
      if (ACCUM)
        Db[idx] += v;
      else
        Db[idx] = v;
    }
  }
}

// ---------------------------------------------------------------------------
// Row softmax over 1024 columns (PAM attention), in place.
// ---------------------------------------------------------------------------
__global__ __launch_bounds__(256) void softmax_rows_kernel(
    float* __restrict__ attn) {
  __shared__ float red[256];
  const int row = blockIdx.x;
  const int b = blockIdx.y;
  float* p = attn + ((size_t)b * 1024 + row) * 1024;
  const int tid = threadIdx.x;
  float v[4];
  float mx = -3.4e38f;
#pragma unroll
  for (int i = 0; i < 4; ++i) {
    v[i] = p[tid + i * 256];
    mx = fmaxf(mx, v[i]);
  }
  red[tid] = mx;
  __syncthreads();
  for (int st = 128; st > 0; st >>= 1) {
    if (tid < st) red[tid] = fmaxf(red[tid], red[tid + st]);
    __syncthreads();
  }
  mx = red[0];
  __syncthreads();
  float s = 0.f;
#pragma unroll
  for (int i = 0; i < 4; ++i) {
    v[i] = __expf(v[i] - mx);
    s += v[i];
  }
  red[tid] = s;
  __syncthreads();
  for (int st = 128; st > 0; st >>= 1) {
    if (tid < st) red[tid] += red[tid + st];
    __syncthreads();
  }
  const float inv = 1.0f / red[0];
#pragma unroll
  for (int i = 0; i < 4; ++i) p[tid + i * 256] = v[i] * inv;
}

// ---------------------------------------------------------------------------
// CAM softmax: a = softmax(rowmax(e) - e) == exp(rowmin(e) - e) / sum.
// ---------------------------------------------------------------------------
__global__ __launch_bounds__(128) void cam_softmax_kernel(
    float* __restrict__ e) {
  __shared__ float red[128];
  const int c = blockIdx.x;
  const int b = blockIdx.y;
  float* p = e + ((size_t)b * 128 + c) * 128;
  const int tid = threadIdx.x;
  float v = p[tid];
  red[tid] = v;
  __syncthreads();
  for (int st = 64; st > 0; st >>= 1) {
    if (tid < st) red[tid] = fminf(red[tid], red[tid + st]);
    __syncthreads();
  }
  const float mn = red[0];
  __syncthreads();
  const float pz = __expf(mn - v);
  red[tid] = pz;
  __syncthreads();
  for (int st = 64; st > 0; st >>= 1) {
    if (tid < st) red[tid] += red[tid + st];
    __syncthreads();
  }
  p[tid] = pz / red[0];
}

// ---------------------------------------------------------------------------
// TIM energy: e[b][t][s] = sum_{c,hw} x[b,c,t,hw] * x[b,c,s,hw]
// ---------------------------------------------------------------------------
__global__ __launch_bounds__(256) void tim_energy_kernel(
    const float* __restrict__ x, float* __restrict__ e) {
  __shared__ float red[256];
  const int pair = blockIdx.x;  // t*8 + s
  const int b = blockIdx.y;
  const int t = pair >> 3;
  const int s = pair & 7;
  const float* xb = x + ((size_t)b << 20);
  float acc = 0.f;
  for (int i = threadIdx.x; i < 131072; i += 256) {
    int c = i >> 10;
    int hw = i & 1023;
    const float* base = xb + (size_t)c * 8192 + hw;
    acc += base[t * 1024] * base[s * 1024];
  }
  red[threadIdx.x] = acc;
  __syncthreads();
  for (int st = 128; st > 0; st >>= 1) {
    if (threadIdx.x < st) red[threadIdx.x] += red[threadIdx.x + st];
    __syncthreads();
  }
  if (threadIdx.x == 0) e[b * 64 + pair] = red[0];
}

__global__ void tim_softmax_kernel(float* __restrict__ e) {
  const int b = blockIdx.x;
  const int t = threadIdx.x;  // 8 threads
  float* r = e + b * 64 + t * 8;
  float mn = r[0];
#pragma unroll
  for (int s = 1; s < 8; ++s) mn = fminf(mn, r[s]);
  float pv[8];
  float sum = 0.f;
#pragma unroll
  for (int s = 0; s < 8; ++s) {
    pv[s] = __expf(mn - r[s]);
    sum += pv[s];
  }
#pragma unroll
  for (int s = 0; s < 8; ++s) r[s] = pv[s] / sum;
}

// out[b,c,t,hw] += g * sum_s a[b][t][s] * x[b,c,s,hw] + x[b,c,t,hw]
__global__ __launch_bounds__(256) void tim_out_kernel(
    const float* __restrict__ x, const float* __restrict__ a,
    const float* __restrict__ g, float* __restrict__ out) {
  const int idx = blockIdx.x * 256 + threadIdx.x;
  const int hw = idx & 1023;
  const int t = (idx >> 10) & 7;
  const int c = (idx >> 13) & 127;
  const int b = idx >> 20;
  const float* xb = x + ((size_t)b << 20);
  const float* ar = a + b * 64 + t * 8;
  float acc = 0.f;
#pragma unroll
  for (int s = 0; s < 8; ++s) acc += ar[s] * xb[(size_t)c * 8192 + s * 1024 + hw];
  out[idx] += g[0] * acc + x[idx];
}

// ---------------------------------------------------------------------------
extern "C" void kernel_launch(void* const* d_in, const int* in_sizes, int n_in,
                              void* d_out, int out_size, void* d_ws,
                              size_t ws_size, hipStream_t stream) {
  const float* x = (const float*)d_in[0];
  const float* Wq = (const float*)d_in[1];
  const float* bq = (const float*)d_in[2];
  const float* Wk = (const float*)d_in[3];
  const float* bk = (const float*)d_in[4];
  const float* Wv = (const float*)d_in[5];
  const float* bv = (const float*)d_in[6];
  const float* gpam = (const float*)d_in[7];
  const float* gcam = (const float*)d_in[8];
  const float* gtim = (const float*)d_in[9];
  float* out = (float*)d_out;

  float* ws = (float*)d_ws;
  float* qf = ws;                   // [8][128][1024]   (1,048,576)
  float* kf = qf + (1 << 20);       // [8][128][1024]   (1,048,576)
  float* vf = kf + (1 << 20);       // [8][1024][1024]  (8,388,608)
  float* attn = vf + (8 << 20);     // [8][1024][1024]  (8,388,608)
  float* cam_e = attn + (8 << 20);  // [8][128][128]    (131,072)
  float* tim_e = cam_e + 131072;    // [8][8][8]        (512)

  // --- QKV projections (WMMA) ---
  proj_kernel<<<dim3(64, 8), 256, 16 * 17 * sizeof(float), stream>>>(
      x, Wq, bq, qf, 16);
  proj_kernel<<<dim3(64, 8), 256, 16 * 17 * sizeof(float), stream>>>(
      x, Wk, bk, kf, 16);
  proj_kernel<<<dim3(512, 8), 256, 16 * 129 * sizeof(float), stream>>>(
      x, Wv, bv, vf, 128);

  // --- PAM: energy = qf^T kf ; softmax rows ; out = gpam * vf attn^T + x ---
  gemm64_kernel<false, false, false><<<dim3(8, 16, 8), 256, 0, stream>>>(
      qf, kf, attn, nullptr, nullptr, /*K=*/128, 1024, 1024, 1024,
      131072, 131072, (size_t)1 << 20, 0);
  softmax_rows_kernel<<<dim3(1024, 8), 256, 0, stream>>>(attn);
  gemm64_kernel<true, true, false><<<dim3(8, 16, 8), 256, 0, stream>>>(
      vf, attn, out, x, gpam, /*K=*/1024, 1024, 1024, 1024,
      (size_t)1 << 20, (size_t)1 << 20, (size_t)1 << 20, (size_t)1 << 20);

  // --- CAM: e = xc xc^T ; negated softmax ; out += gcam * a xc + x ---
  gemm64_kernel<true, true, false><<<dim3(1, 2, 8), 256, 0, stream>>>(
      x, x, cam_e, nullptr, nullptr, /*K=*/8192, 8192, 8192, 128,
      (size_t)1 << 20, (size_t)1 << 20, 16384, 0);
  cam_softmax_kernel<<<dim3(128, 8), 128, 0, stream>>>(cam_e);
  gemm64_kernel<true, false, true><<<dim3(64, 2, 8), 256, 0, stream>>>(
      cam_e, x, out, x, gcam, /*K=*/128, 128, 8192, 8192,
      16384, (size_t)1 << 20, (size_t)1 << 20, (size_t)1 << 20);

  // --- TIM: tiny 8x8 attention, memory-bound ---
  tim_energy_kernel<<<dim3(64, 8), 256, 0, stream>>>(x, tim_e);
  tim_softmax_kernel<<<8, 8, 0, stream>>>(tim_e);
  tim_out_kernel<<<32768, 256, 0, stream>>>(x, tim_e, gtim, out);
}